// MemoryCell_28260884807887
// MI455X (gfx1250) — compile-verified
//
#include <hip/hip_runtime.h>
#include <stdint.h>

// ---------------------------------------------------------------------------
// MemoryCell scan for MI455X (gfx1250).
//   grid:  B/64 blocks, 64 threads; thread owns one batch row's (x,y) state.
//   Pipeline per 32-step time chunk:
//     async DMA (global->LDS, b128, coalesced)  ->  s_wait_asynccnt  ->
//     barrier -> 32 sequential state updates from LDS -> barrier ->
//     coalesced b128 store of the x-trace tile -> barrier.
//   Input tile double-buffered so chunk c+1's DMA overlaps chunk c's compute.
// ---------------------------------------------------------------------------

#define BLOCK  64
#define TCH    32
#define T_TOT  768
#define NCH    (T_TOT / TCH)        // 24 chunks
#define INROW  (TCH * 2 + 2)        // 66 floats; 264 B stride (8B aligned, bank-spread)
#define OUTROW (TCH + 4)            // 36 floats; 144 B stride (16B aligned)

struct SynParams { const float* p[26]; };

__device__ __forceinline__ float sigmoidf(float z) {
    return 1.0f / (1.0f + __expf(-z));
}

// Per-lane async DMA: 16 bytes global -> LDS, tracked by ASYNCcnt.
__device__ __forceinline__ void async_ld16(uint32_t lds_off, const float* g) {
    asm volatile("global_load_async_to_lds_b128 %0, %1, off"
                 :: "v"(lds_off), "v"(g)
                 : "memory");
}

#if __has_builtin(__builtin_amdgcn_s_wait_asynccnt)
#define WAIT_ASYNCCNT(N) __builtin_amdgcn_s_wait_asynccnt(N)
#else
#define WAIT_ASYNCCNT(N) asm volatile("s_wait_asynccnt %0" :: "i"(N) : "memory")
#endif

__global__ __launch_bounds__(BLOCK)
void memcell_scan_kernel(const float* __restrict__ in, float* __restrict__ out,
                         SynParams P) {
    __shared__ float lds_in[2][BLOCK * INROW];   // 33792 B (ping-pong input tiles)
    __shared__ float lds_out[BLOCK * OUTROW];    //  9216 B (x-trace tile)

    const int tid = threadIdx.x;
    const int b0  = blockIdx.x * BLOCK;

    // ---- uniform scalar parameters (addresses are SGPR kernel args) -------
    const float capx = P.p[0][0], capy = P.p[1][0];
    const float g_ax = P.p[2][0],  m_ax = P.p[3][0],  s_ax = P.p[4][0],  p_ax = P.p[5][0];
    const float g_by = P.p[6][0],  m_by = P.p[7][0],  s_by = P.p[8][0],  p_by = P.p[9][0];
    const float g_xx = P.p[10][0], m_xx = P.p[11][0], s_xx = P.p[12][0], p_xx = P.p[13][0];
    const float g_xy = P.p[14][0], m_xy = P.p[15][0], s_xy = P.p[16][0], p_xy = P.p[17][0];
    const float g_yx = P.p[18][0], m_yx = P.p[19][0], s_yx = P.p[20][0], p_yx = P.p[21][0];
    const float g_yy = P.p[22][0], m_yy = P.p[23][0], s_yy = P.p[24][0], p_yy = P.p[25][0];

    const float icx = 1.0f / capx, icy = 1.0f / capy;
    const float Gax = g_ax * icx, Gxx = g_xx * icx, Gyx = g_yx * icx;
    const float Gby = g_by * icy, Gyy = g_yy * icy, Gxy = g_xy * icy;

    // Issue one chunk's DMA: 64 rows x 16 float4 (=32 timesteps x 2ch).
    // Lanes 0..15 / 16..31 each cover a full row -> 256 B contiguous/16 lanes.
    auto issueLoads = [&](int chunk, int buf) {
        const float* gbase = in + (size_t)b0 * (T_TOT * 2) + (size_t)chunk * (TCH * 2);
        const uint32_t lbase = (uint32_t)(uintptr_t)(&lds_in[buf][0]);
#pragma unroll
        for (int k = 0; k < 16; ++k) {
            const int f = k * BLOCK + tid;           // 0..1023 float4 ids
            const int r = f >> 4;                    // row (batch-local)
            const int j = f & 15;                    // float4 within row
            async_ld16(lbase + (uint32_t)(r * (INROW * 4) + j * 16),
                       gbase + (size_t)r * (T_TOT * 2) + j * 4);
        }
    };

    float x = 0.0f, y = 0.0f;

    issueLoads(0, 0);

    for (int c = 0; c < NCH; ++c) {
        if (c + 1 < NCH) {
            issueLoads(c + 1, (c + 1) & 1);          // overlap DMA with compute
            WAIT_ASYNCCNT(16);                       // oldest 16 (= chunk c) done
        } else {
            WAIT_ASYNCCNT(0);
        }
        __syncthreads();                             // tile c visible block-wide

        // ---- sequential recurrence over this chunk ------------------------
        const float* row  = &lds_in[c & 1][tid * INROW];
        float*       orow = &lds_out[tid * OUTROW];
#pragma unroll 8
        for (int t = 0; t < TCH; ++t) {
            const float a = row[2 * t + 0];
            const float b = row[2 * t + 1];
            const float ix = Gax * sigmoidf(s_ax * (a - m_ax)) * (p_ax - x)
                           + Gxx * sigmoidf(s_xx * (x - m_xx)) * (p_xx - x)
                           + Gyx * sigmoidf(s_yx * (y - m_yx)) * (p_yx - x);
            const float iy = Gby * sigmoidf(s_by * (b - m_by)) * (p_by - y)
                           + Gyy * sigmoidf(s_yy * (y - m_yy)) * (p_yy - y)
                           + Gxy * sigmoidf(s_xy * (x - m_xy)) * (p_xy - y);
            x += ix;
            y += iy;
            orow[t] = x;
        }
        __syncthreads();                             // out tile complete

        // ---- coalesced flush of the x-trace tile --------------------------
        float* obase = out + (size_t)b0 * T_TOT + (size_t)c * TCH;
#pragma unroll
        for (int k = 0; k < 8; ++k) {
            const int f = k * BLOCK + tid;           // 0..511 float4 ids
            const int r = f >> 3;
            const int j = f & 7;
            const float4 v = *(const float4*)&lds_out[r * OUTROW + j * 4];
            *(float4*)(obase + (size_t)r * T_TOT + j * 4) = v;
        }
        __syncthreads();                             // lds_out reusable next chunk
    }
}

extern "C" void kernel_launch(void* const* d_in, const int* in_sizes, int n_in,
                              void* d_out, int out_size, void* d_ws, size_t ws_size,
                              hipStream_t stream) {
    (void)n_in; (void)out_size; (void)d_ws; (void)ws_size;
    const float* in  = (const float*)d_in[0];
    float*       out = (float*)d_out;

    SynParams P;
    for (int i = 0; i < 26; ++i) P.p[i] = (const float*)d_in[1 + i];

    const int B = in_sizes[0] / (T_TOT * 2);         // 16384
    dim3 grid(B / BLOCK), block(BLOCK);
    hipLaunchKernelGGL(memcell_scan_kernel, grid, block, 0, stream, in, out, P);
}